// AxialAttention_71734543778490
// MI455X (gfx1250) — compile-verified
//
#include <hip/hip_runtime.h>
#include <hip/hip_bf16.h>
#include <math.h>

typedef float v2f __attribute__((ext_vector_type(2)));
typedef float v8f __attribute__((ext_vector_type(8)));

#define NB 32
#define HH 56
#define WW 56
#define CC 128
#define OUTC 128
#define GG 8
#define GC 16
#define GC2 8
#define HW 3136            // H*W
// workspace offsets (in floats)
#define OFF_WCAT  0u            // 128*256 = 32768 (pair-swizzled [k/2][n][2])
#define OFF_BIAS  32768u        // 256
#define OFF_QEMB  33024u        // 56*56*8 = 25088
#define OFF_KEMB  58112u        // 25088
#define OFF_VEMB  83200u        // 56*56*16 = 50176
#define OFF_QT    133376u       // 32*56*8*56*8 = 6422528
#define OFF_KT    6555904u      // 6422528
#define OFF_VT    12978432u     // 32*56*8*56*16 = 12845056
#define OFF_S     25823488u     // 32*8*56*56*56 = 44957696

__device__ __forceinline__ v8f wmma_f32(v2f a, v2f b, v8f c) {
  // D = A(16x4) * B(4x16) + C, full fp32 (V_WMMA_F32_16X16X4_F32)
  return __builtin_amdgcn_wmma_f32_16x16x4_f32(false, a, false, b, (short)0, c, false, false);
}

// ---------------- Kernel 0: fold BN into weights, build relative embeddings ----
__global__ __launch_bounds__(256) void prep_kernel(
    const float* __restrict__ wq, const float* __restrict__ wk, const float* __restrict__ wv,
    const float* __restrict__ qrel, const float* __restrict__ krel, const float* __restrict__ vrel,
    const float* __restrict__ gq, const float* __restrict__ bq,
    const float* __restrict__ gk, const float* __restrict__ bk,
    const float* __restrict__ gv, const float* __restrict__ bv,
    float* __restrict__ ws) {
  int idx = blockIdx.x * blockDim.x + threadIdx.x;
  const float invs = rsqrtf(1.0f + 1e-3f);
  if (idx < 32768) {               // Wcat pair-swizzled: [(k>>1)][n][k&1]
    int kk = idx >> 8, n = idx & 255;
    float val;
    if (n < 64)        val = wq[kk * 64 + n]          * (gq[n]       * invs);
    else if (n < 128)  val = wk[kk * 64 + (n - 64)]   * (gk[n - 64]  * invs);
    else               val = wv[kk * 128 + (n - 128)] * (gv[n - 128] * invs);
    ws[OFF_WCAT + (((kk >> 1) * 256 + n) << 1) + (kk & 1)] = val;
    return;
  }
  idx -= 32768;
  if (idx < 256) {                 // concatenated bias
    ws[OFF_BIAS + idx] = (idx < 64) ? bq[idx] : (idx < 128 ? bk[idx - 64] : bv[idx - 128]);
    return;
  }
  idx -= 256;
  if (idx < 25088) {               // q_emb[i][j][c] = q_rel[i-j+55][c]
    int i = idx / 448, r = idx % 448, j = r >> 3, c = r & 7;
    ws[OFF_QEMB + idx] = qrel[(i - j + 55) * 8 + c];
    return;
  }
  idx -= 25088;
  if (idx < 25088) {
    int i = idx / 448, r = idx % 448, j = r >> 3, c = r & 7;
    ws[OFF_KEMB + idx] = krel[(i - j + 55) * 8 + c];
    return;
  }
  idx -= 25088;
  if (idx < 50176) {               // v_emb[i][j][c] = v_rel[i-j+55][c]
    int i = idx / 896, r = idx % 896, j = r >> 4, c = r & 15;
    ws[OFF_VEMB + idx] = vrel[(i - j + 55) * 16 + c];
    return;
  }
}

// ---------------- Kernel 1: fused QKV GEMM (WMMA f32 16x16x4) + BN epilogue ----
// grid.x = M/16 = 6272 tiles; 256 threads = 8 waves, each wave 2 N-tiles (covers N=256)
__global__ __launch_bounds__(256) void qkv_gemm_kernel(const float* __restrict__ x,
                                                       float* __restrict__ ws) {
  __shared__ float As[2048];               // pair-swizzled: [(k>>1)][m][k&1]
  const int tid = threadIdx.x, mt = blockIdx.x;
  for (int idx = tid; idx < 2048; idx += 256) {
    int row = idx >> 7, k = idx & 127;
    As[(((k >> 1) << 4) + row << 1) + (k & 1)] = x[mt * 2048 + idx];
  }
  __syncthreads();
  const float* __restrict__ Wc   = ws + OFF_WCAT;
  const float* __restrict__ bias = ws + OFF_BIAS;
  const int wave = tid >> 5, lane = tid & 31, half = lane >> 4, l16 = lane & 15;
  const int n0 = wave * 32 + l16;
  v8f acc0 = {}; v8f acc1 = {};
  for (int k0 = 0; k0 < 128; k0 += 4) {
    int kp = (k0 >> 1) + half;             // K-pair index (kb = 2*kp)
    v2f a  = *(const v2f*)&As[((kp << 4) + l16) << 1];
    v2f b0 = *(const v2f*)&Wc[(kp * 256 + n0) << 1];
    v2f b1 = *(const v2f*)&Wc[(kp * 256 + n0 + 16) << 1];
    acc0 = wmma_f32(a, b0, acc0);
    acc1 = wmma_f32(a, b1, acc1);
  }
  float* __restrict__ Qt = ws + OFF_QT;
  float* __restrict__ Kt = ws + OFF_KT;
  float* __restrict__ Vt = ws + OFF_VT;
  #pragma unroll
  for (int r = 0; r < 8; ++r) {
    int m  = mt * 16 + r + 8 * half;
    int b_ = m / HW, rem = m - b_ * HW;
    int i  = rem / WW, w = rem - i * WW;
    #pragma unroll
    for (int t = 0; t < 2; ++t) {
      int n = n0 + t * 16;
      float val = (t == 0 ? acc0[r] : acc1[r]) + bias[n];
      if (n < 64) {
        int g = n >> 3, c = n & 7;
        Qt[(((b_ * WW + w) * GG + g) * HH + i) * GC2 + c] = val;
      } else if (n < 128) {
        int nn = n - 64; int g = nn >> 3, c = nn & 7;
        Kt[(((b_ * WW + w) * GG + g) * HH + i) * GC2 + c] = val;
      } else {
        int nn = n - 128; int g = nn >> 4, c = nn & 15;
        Vt[(((b_ * WW + w) * GG + g) * HH + i) * GC + c] = val;
      }
    }
  }
}

// ---------------- Kernel 2: S[b,g,i,j,w] = BN(qk) + BN(qr) + BN(kr^T) ----------
// grid = (G, W, N); block = 256 = 8 waves; each wave does 2 of 16 (i,j) 16x16 tiles
// Qs/Ks pair-swizzled over c: [(c>>1)][row][c&1]
__global__ __launch_bounds__(256) void scores_kernel(
    const float* __restrict__ gqk_p, const float* __restrict__ bqk_p,
    const float* __restrict__ gqr_p, const float* __restrict__ bqr_p,
    const float* __restrict__ gkr_p, const float* __restrict__ bkr_p,
    float* __restrict__ ws) {
  __shared__ float Qs[448], Ks[448];
  const int g = blockIdx.x, w = blockIdx.y, b = blockIdx.z, tid = threadIdx.x;
  const float* __restrict__ Qt = ws + OFF_QT;
  const float* __restrict__ Kt = ws + OFF_KT;
  const int base = ((b * WW + w) * GG + g) * 448;
  for (int idx = tid; idx < 896; idx += 256) {
    int which = idx >= 448, e = idx - which * 448;
    int row = e >> 3, c = e & 7;
    int sw = (((c >> 1) * 56 + row) << 1) + (c & 1);
    if (which) Ks[sw] = Kt[base + e];
    else       Qs[sw] = Qt[base + e];
  }
  __syncthreads();
  const float invs = rsqrtf(1.0f + 1e-3f);
  const float gqk = gqk_p[g] * invs, gqr = gqr_p[g] * invs, gkr = gkr_p[g] * invs;
  const float bsum = bqk_p[g] + bqr_p[g] + bkr_p[g];
  const float* __restrict__ qemb = ws + OFF_QEMB;
  const float* __restrict__ kemb = ws + OFF_KEMB;
  float* __restrict__ S = ws + OFF_S;
  const int wave = tid >> 5, lane = tid & 31, half = lane >> 4, l16 = lane & 15;
  const int sb = (b * GG + g) * (HH * HH);     // S tile base / W
  const v2f zero2 = {0.f, 0.f};
  #pragma unroll
  for (int tt = 0; tt < 2; ++tt) {
    const int t = wave * 2 + tt, ti = t >> 2, tj = t & 3;
    const int ii = ti * 16 + l16, jj = tj * 16 + l16;
    v8f acc = {};
    #pragma unroll
    for (int kk0 = 0; kk0 < 8; kk0 += 4) {     // K(reduction) = gc/2 = 8
      int kp = (kk0 >> 1) + half;              // c-pair index
      v2f a  = (ii < 56) ? *(const v2f*)&Qs[(kp * 56 + ii) << 1] : zero2;
      v2f bb = (jj < 56) ? *(const v2f*)&Ks[(kp * 56 + jj) << 1] : zero2;
      acc = wmma_f32(a, bb, acc);
    }
    #pragma unroll
    for (int r = 0; r < 8; ++r) {
      int i = ti * 16 + r + 8 * half;
      int j = tj * 16 + l16;
      if (i < 56 && j < 56) {
        float qr = 0.f, kr = 0.f;
        #pragma unroll
        for (int c = 0; c < 8; ++c) {
          int swi = (((c >> 1) * 56 + i) << 1) + (c & 1);
          int swj = (((c >> 1) * 56 + j) << 1) + (c & 1);
          qr += Qs[swi] * qemb[(i * 56 + j) * 8 + c];        // q_emb[i][j][c]
          kr += Ks[swj] * kemb[(j * 56 + i) * 8 + c];        // k_emb[j][i][c] (transposed kr)
        }
        S[(size_t)(sb + i * 56 + j) * WW + w] = gqk * acc[r] + gqr * qr + gkr * kr + bsum;
      }
    }
  }
}

// ---------------- Kernel 3: softmax over w (last axis, contiguous) -------------
__global__ __launch_bounds__(256) void softmax_kernel(float* __restrict__ ws) {
  float* __restrict__ S = ws + OFF_S;
  const int tid = threadIdx.x, wave = tid >> 5, lane = tid & 31;
  const long long row = (long long)blockIdx.x * 8 + wave;   // N*G*H*H rows of 56
  float* p = S + row * 56;
  float v0 = p[lane];
  float v1 = (lane + 32 < 56) ? p[lane + 32] : -3.0e38f;
  float m = fmaxf(v0, v1);
  #pragma unroll
  for (int off = 16; off > 0; off >>= 1) m = fmaxf(m, __shfl_xor(m, off, 32));
  float e0 = __expf(v0 - m);
  float e1 = (lane + 32 < 56) ? __expf(v1 - m) : 0.f;
  float s = e0 + e1;
  #pragma unroll
  for (int off = 16; off > 0; off >>= 1) s += __shfl_xor(s, off, 32);
  float inv = 1.f / s;
  p[lane] = e0 * inv;
  if (lane + 32 < 56) p[lane + 32] = e1 * inv;
}

// ---------------- Kernel 4: out = BN(sim @ V) + BN(sim ⊙ v_emb) ----------------
// grid = (G, W, N); block = 128 = 4 waves; wave ti handles rows i = 16*ti..16*ti+15
// sims pair-swizzled over j: [(j>>1)][i][j&1]; Vs pair-swizzled over j: [(j>>1)][c][j&1]
__global__ __launch_bounds__(128) void out_kernel(
    const float* __restrict__ gsv_p, const float* __restrict__ bsv_p,
    const float* __restrict__ gsve_p, const float* __restrict__ bsve_p,
    const float* __restrict__ ws, float* __restrict__ out) {
  __shared__ float sims[3136];
  __shared__ float Vs[896];
  const int g = blockIdx.x, w = blockIdx.y, b = blockIdx.z, tid = threadIdx.x;
  const float* __restrict__ S    = ws + OFF_S;
  const float* __restrict__ Vt   = ws + OFF_VT;
  const float* __restrict__ vemb = ws + OFF_VEMB;
  const size_t sbase = (size_t)(b * GG + g) * (HH * HH) * WW;
  for (int idx = tid; idx < 3136; idx += 128) {
    int i = idx / 56, j = idx - i * 56;
    sims[(((j >> 1) * 56 + i) << 1) + (j & 1)] = S[sbase + (size_t)idx * WW + w];
  }
  const int vbase = ((b * WW + w) * GG + g) * 896;
  for (int idx = tid; idx < 896; idx += 128) {
    int j = idx >> 4, c = idx & 15;
    Vs[(((j >> 1) << 4) + c << 1) + (j & 1)] = Vt[vbase + idx];
  }
  __syncthreads();
  const int wave = tid >> 5, lane = tid & 31, half = lane >> 4, l16 = lane & 15;
  const int ti = wave;
  const float invs = rsqrtf(1.0f + 1e-3f);
  const int ch = g * 16 + l16;
  const float gsv = gsv_p[ch] * invs, bsv = bsv_p[ch];
  const float gsve = gsve_p[ch] * invs, bsve = bsve_p[ch];
  v8f acc = {};
  {
    const int ii = ti * 16 + l16;
    const int iic = ii < 56 ? ii : 55;     // pad rows read valid LDS; discarded at store
    for (int j0 = 0; j0 < 56; j0 += 4) {   // K(reduction) = H = 56
      int kp = (j0 >> 1) + half;           // j-pair index
      v2f a  = *(const v2f*)&sims[(kp * 56 + iic) << 1];
      v2f bb = *(const v2f*)&Vs[((kp << 4) + l16) << 1];
      acc = wmma_f32(a, bb, acc);
    }
  }
  #pragma unroll
  for (int r = 0; r < 8; ++r) {
    int i = ti * 16 + r + 8 * half;
    if (i < 56) {
      float sve = 0.f;
      for (int j = 0; j < 56; ++j) {
        float sv_ij = sims[(((j >> 1) * 56 + i) << 1) + (j & 1)];
        sve += sv_ij * vemb[(j * 56 + i) * 16 + l16];        // v_emb[j][i][c]
      }
      out[((b * HH + i) * WW + w) * OUTC + ch] = gsv * acc[r] + bsv + gsve * sve + bsve;
    }
  }
}

extern "C" void kernel_launch(void* const* d_in, const int* in_sizes, int n_in,
                              void* d_out, int out_size, void* d_ws, size_t ws_size,
                              hipStream_t stream) {
  (void)in_sizes; (void)n_in; (void)out_size; (void)ws_size;
  const float* x     = (const float*)d_in[0];
  const float* w_q   = (const float*)d_in[1];
  const float* w_k   = (const float*)d_in[2];
  const float* w_v   = (const float*)d_in[3];
  const float* q_rel = (const float*)d_in[4];
  const float* k_rel = (const float*)d_in[5];
  const float* v_rel = (const float*)d_in[6];
  const float* g_q = (const float*)d_in[7],  * b_q = (const float*)d_in[8];
  const float* g_k = (const float*)d_in[9],  * b_k = (const float*)d_in[10];
  const float* g_v = (const float*)d_in[11], * b_v = (const float*)d_in[12];
  const float* g_qk = (const float*)d_in[13], * b_qk = (const float*)d_in[14];
  const float* g_qr = (const float*)d_in[15], * b_qr = (const float*)d_in[16];
  const float* g_kr = (const float*)d_in[17], * b_kr = (const float*)d_in[18];
  const float* g_sv = (const float*)d_in[19], * b_sv = (const float*)d_in[20];
  const float* g_sve = (const float*)d_in[21], * b_sve = (const float*)d_in[22];
  float* ws  = (float*)d_ws;
  float* out = (float*)d_out;

  prep_kernel<<<(133376 + 255) / 256, 256, 0, stream>>>(
      w_q, w_k, w_v, q_rel, k_rel, v_rel, g_q, b_q, g_k, b_k, g_v, b_v, ws);
  qkv_gemm_kernel<<<6272, 256, 0, stream>>>(x, ws);
  scores_kernel<<<dim3(GG, WW, NB), 256, 0, stream>>>(
      g_qk, b_qk, g_qr, b_qr, g_kr, b_kr, ws);
  softmax_kernel<<<(NB * GG * HH * HH) / 8, 256, 0, stream>>>(ws);
  out_kernel<<<dim3(GG, WW, NB), 128, 0, stream>>>(
      g_sv, b_sv, g_sve, b_sve, ws, out);
}